// VectorQuantizer1_76261439307887
// MI455X (gfx1250) — compile-verified
//
#include <hip/hip_runtime.h>
#include <hip/hip_bf16.h>
#include <math.h>

typedef __attribute__((ext_vector_type(2))) float v2f;
typedef __attribute__((ext_vector_type(4))) float v4f;
typedef __attribute__((ext_vector_type(8))) float v8f;

#define EDIM 256
#define NE 1024
#define HW 1024                 // 32*32 positions per batch
#define BATCH_STRIDE (EDIM*HW)  // 262144
#define TOT_ELEMS 8388608       // 32*256*32*32

// d_out layout (floats): [loss][z_q 8388608][perplexity][idx 32768][z_q1 8388608]
#define ZQ_OFF   1
#define PERP_OFF 8388609
#define IDX_OFF  8388610
#define ZQ1_OFF  8421378

#define WMMA_F32(A, B, C) \
    __builtin_amdgcn_wmma_f32_16x16x4_f32(false, (A), false, (B), (short)0, (C), false, false)

// -------------------- init: zero hist + loss accumulator --------------------
__global__ void vq_init(int* __restrict__ hist, float* __restrict__ lossacc) {
    int t = blockIdx.x * blockDim.x + threadIdx.x;
    if (t < NE) hist[t] = 0;
    if (t == 0) *lossacc = 0.0f;
}

// -------------------- prep: pack B fragments + embN[j] = ||e_j||^2 --------------------
// Bpack layout: for code j, channel c with kk=c>>2, half=(c>>1)&1, i=c&1:
//   Bpack[kk*4096 + half*2048 + j*2 + i] = emb[j][c]
// -> one aligned b64 per B fragment in the GEMM (v0=K4kk+2h, v1=K4kk+2h+1).
__global__ void vq_prep(const float* __restrict__ emb, float* __restrict__ Bpack,
                        float* __restrict__ embN) {
    int j = blockIdx.x * blockDim.x + threadIdx.x;
    if (j >= NE) return;
    float norm = 0.0f;
    for (int c = 0; c < EDIM; ++c) {
        float v = emb[j * EDIM + c];
        Bpack[(c >> 2) * 4096 + ((c >> 1) & 1) * 2048 + j * 2 + (c & 1)] = v;
        norm = fmaf(v, v, norm);
    }
    embN[j] = norm;
}

// -------------------- main: fp32 WMMA distance GEMM + fused argmin --------------------
// 64 threads = 2 waves per block; each wave owns 32 token rows (two 16-row A tiles).
// Per kp iteration: 2 ds_load_b128 (A frags for 2 k-steps, both tiles) +
// 2 global_load_b64 (shared B frags) + 4 v_wmma (4 independent chains).
__global__ __launch_bounds__(64) void vq_argmin(
        const float* __restrict__ z, const float* __restrict__ Bpack,
        const float* __restrict__ embN, int* __restrict__ idxbuf,
        int* __restrict__ hist, float* __restrict__ out) {
    __shared__ float smem[16384];            // 2 waves * 32 rows * 256 ch = 64 KB
    const int lane = threadIdx.x & 31;
    const int wib  = threadIdx.x >> 5;       // wave in block: 0..1
    const int wave = blockIdx.x * 2 + wib;   // 0..1023
    const int row_base = wave * 32;
    const size_t zb = (size_t)(row_base >> 10) * BATCH_STRIDE + (row_base & 1023);

    // Stage A: 32 tokens x 256 channels, packed per 16-row tile as
    //   [(kp*2+h)*64 + r*4 + s*2 + i]  with channel c = 8*kp + 4*s + 2*h + i
    // so one ds_load_b128 yields A fragments for k-steps 2kp (.xy) and 2kp+1 (.zw).
    for (int t = lane; t < 8192; t += 32) {
        int c = t >> 5, r32 = t & 31;        // global: 128B contiguous per iter
        int tile = r32 >> 4, r = r32 & 15;
        int kp = c >> 3, s = (c >> 2) & 1, h = (c >> 1) & 1, i = c & 1;
        smem[wib * 8192 + tile * 4096 + (kp * 2 + h) * 64 + r * 4 + s * 2 + i] =
            z[zb + (size_t)c * HW + r32];
    }
    __syncthreads();

    const int half = lane >> 4;              // K-half / row-half selector
    const int l15  = lane & 15;
    const int aoff0 = wib * 8192 + half * 64 + l15 * 4;   // + kp*128  (tile0)
    const int aoff1 = aoff0 + 4096;                       // tile1
    const float* bbase = Bpack + half * 2048 + l15 * 2;   // + j0*2 + kk*4096

    float bestv[2][8];
    int   besti[2][8];
#pragma unroll
    for (int t = 0; t < 2; ++t)
#pragma unroll
        for (int e = 0; e < 8; ++e) { bestv[t][e] = 3.4e38f; besti[t][e] = 0; }

    for (int jt = 0; jt < 64; ++jt) {        // 64 column tiles of 16 codes
        const int j0 = jt * 16;
        const float* bp = bbase + j0 * 2;
        v8f acc0e = {}, acc0o = {}, acc1e = {}, acc1o = {};
#pragma unroll 4
        for (int kp = 0; kp < 32; ++kp) {    // two K=4 steps per iteration
            v4f a0 = *(const v4f*)&smem[aoff0 + kp * 128];   // ds_load_b128
            v4f a1 = *(const v4f*)&smem[aoff1 + kp * 128];
            v2f b0 = *(const v2f*)(bp + (2 * kp)     * 4096); // global_load_b64
            v2f b1 = *(const v2f*)(bp + (2 * kp + 1) * 4096);
            acc0e = WMMA_F32(a0.xy, b0, acc0e);
            acc1e = WMMA_F32(a1.xy, b0, acc1e);
            acc0o = WMMA_F32(a0.zw, b1, acc0o);
            acc1o = WMMA_F32(a1.zw, b1, acc1o);
        }
        const float en  = embN[j0 + l15];
        const int   col = j0 + l15;
#pragma unroll
        for (int e = 0; e < 8; ++e) {
            float s0 = acc0e[e] + acc0o[e];
            float s1 = acc1e[e] + acc1o[e];
            float d0 = fmaf(-2.0f, s0, en);  // ||e||^2 - 2 z.e (||z||^2 const per row)
            float d1 = fmaf(-2.0f, s1, en);
            if (d0 < bestv[0][e]) { bestv[0][e] = d0; besti[0][e] = col; }
            if (d1 < bestv[1][e]) { bestv[1][e] = d1; besti[1][e] = col; }
        }
    }

    // Reduce min across the 16 lanes of each half (first-index tie-break like argmin)
#pragma unroll
    for (int m = 1; m < 16; m <<= 1) {
#pragma unroll
        for (int t = 0; t < 2; ++t)
#pragma unroll
            for (int e = 0; e < 8; ++e) {
                float ov = __shfl_xor(bestv[t][e], m, 32);
                int   oi = __shfl_xor(besti[t][e], m, 32);
                if (ov < bestv[t][e] || (ov == bestv[t][e] && oi < besti[t][e])) {
                    bestv[t][e] = ov; besti[t][e] = oi;
                }
            }
    }

    if (l15 == 0) {                      // lane 0 -> rows 0-7, lane 16 -> rows 8-15
#pragma unroll
        for (int t = 0; t < 2; ++t)
#pragma unroll
            for (int e = 0; e < 8; ++e) {
                int row = row_base + t * 16 + half * 8 + e;
                int k = besti[t][e];
                idxbuf[row] = k;
                out[IDX_OFF + row] = (float)k;
                atomicAdd(&hist[k], 1);
            }
    }
}

// -------------------- outputs: gather codes, write z_q/z_q1, loss partial sums --------------------
__global__ void vq_outputs(const float* __restrict__ z, const float* __restrict__ emb,
                           const int* __restrict__ idxbuf, float* __restrict__ out,
                           float* __restrict__ lossacc) {
    __shared__ float partial[8];
    int t   = blockIdx.x * 256 + threadIdx.x;     // (b,c,hw) flat, matches z layout
    int b   = t >> 18;
    int rem = t & (BATCH_STRIDE - 1);
    int c   = rem >> 10;
    int hw  = rem & 1023;
    int n   = (b << 10) | hw;
    int k   = idxbuf[n];
    float e  = emb[k * EDIM + c];
    float zv = z[t];
    out[ZQ_OFF  + t] = e;                          // z_q  (straight-through == z_q1)
    out[ZQ1_OFF + t] = e;                          // z_q1
    float diff = e - zv;
    float sq = diff * diff;
#pragma unroll
    for (int m = 16; m >= 1; m >>= 1) sq += __shfl_xor(sq, m, 32);
    int lane = threadIdx.x & 31, w = threadIdx.x >> 5;
    if (lane == 0) partial[w] = sq;
    __syncthreads();
    if (threadIdx.x == 0) {
        float s = 0.0f;
        for (int i = 0; i < 8; ++i) s += partial[i];
        atomicAdd(lossacc, s);
    }
}

// -------------------- finalize: loss scalar + perplexity --------------------
__global__ void vq_finalize(const int* __restrict__ hist,
                            const float* __restrict__ lossacc,
                            float* __restrict__ out) {
    __shared__ float partial[32];
    int t = threadIdx.x;                           // 1024 threads, 1 block
    float p = (float)hist[t] * (1.0f / 32768.0f);
    float term = p * logf(p + 1e-10f);
#pragma unroll
    for (int m = 16; m >= 1; m >>= 1) term += __shfl_xor(term, m, 32);
    int lane = t & 31, w = t >> 5;
    if (lane == 0) partial[w] = term;
    __syncthreads();
    if (t == 0) {
        float s = 0.0f;
        for (int i = 0; i < 32; ++i) s += partial[i];
        out[PERP_OFF] = expf(-s);
        // loss = mean(sq) + BETA*mean(sq) = 1.25 * sum / 8388608
        out[0] = lossacc[0] * (1.25f / 8388608.0f);
    }
}

extern "C" void kernel_launch(void* const* d_in, const int* in_sizes, int n_in,
                              void* d_out, int out_size, void* d_ws, size_t ws_size,
                              hipStream_t stream) {
    (void)in_sizes; (void)n_in; (void)out_size; (void)ws_size;
    const float* z   = (const float*)d_in[0];   // (32,256,32,32)
    const float* emb = (const float*)d_in[1];   // (1024,256)
    float* out = (float*)d_out;
    float* ws  = (float*)d_ws;
    // workspace layout (float-index units)
    float* Bpack   = ws;                        // 262144
    float* embN    = ws + 262144;               // 1024
    int*   idxbuf  = (int*)(ws + 263168);       // 32768
    int*   hist    = (int*)(ws + 295936);       // 1024
    float* lossacc = ws + 296960;               // 1

    vq_init    <<<4,   256, 0, stream>>>(hist, lossacc);
    vq_prep    <<<4,   256, 0, stream>>>(emb, Bpack, embN);
    vq_argmin  <<<512,  64, 0, stream>>>(z, Bpack, embN, idxbuf, hist, out);
    vq_outputs <<<TOT_ELEMS / 256, 256, 0, stream>>>(z, emb, idxbuf, out, lossacc);
    vq_finalize<<<1,  1024, 0, stream>>>(hist, lossacc, out);
}